// MCLoss_29197187678935
// MI455X (gfx1250) — compile-verified
//
#include <hip/hip_runtime.h>
#include <math.h>

typedef __attribute__((ext_vector_type(2))) float v2f;
typedef __attribute__((ext_vector_type(8))) float v8f;

#define RB 48  // row length in floats for the transposed diff table (B=16 * 3)

// ---------------------------------------------------------------------------
// Wave32 sum-reduction via V_WMMA_F32_16X16X4_F32 (see round-1 derivation).
// Requires EXEC == all ones at the call site.
// ---------------------------------------------------------------------------
__device__ __forceinline__ float wave_sum_wmma(float s) {
    v2f a; a[0] = s;    a[1] = 0.0f;
    v2f b; b[0] = 1.0f; b[1] = 1.0f;
    v8f c = {};
    c = __builtin_amdgcn_wmma_f32_16x16x4_f32(
            /*neg_a=*/false, a, /*neg_b=*/false, b,
            /*c_mod=*/(short)0, c, /*reuse_a=*/false, /*reuse_b=*/false);
    float t = ((c[0] + c[1]) + (c[2] + c[3])) + ((c[4] + c[5]) + (c[6] + c[7]));
    t += __shfl_xor(t, 16, 32);
    return t;
}

__device__ __forceinline__ void block_store_partial(float s, float* dst) {
    const float w = wave_sum_wmma(s);
    __shared__ float wsum[8];
    const int lane = threadIdx.x & 31;
    const int wid  = threadIdx.x >> 5;
    if (lane == 0) wsum[wid] = w;
    __syncthreads();
    if (threadIdx.x == 0) {
        const int nw = (int)(blockDim.x >> 5);
        float t = 0.0f;
        for (int i = 0; i < nw; ++i) t += wsum[i];
        *dst = t;
    }
}

// ---------------------------------------------------------------------------
// Pass 0 (specialized B=16): build transposed diff table dT[(N+1) x 48],
// dT[n][b*3+c] = gt[b][n][c] - pr[b][n][c];  row n==N is all zeros (pad row).
// Reads are lane-coalesced per b-iteration; each thread writes its own
// contiguous 192B row (12 x b128 stores, every sector fully covered).
// ---------------------------------------------------------------------------
__global__ void mcl_difft_kernel(const float* __restrict__ g,
                                 const float* __restrict__ p,
                                 float* __restrict__ dT, int N) {
    const int n = blockIdx.x * blockDim.x + threadIdx.x;
    if (n > N) return;                      // N+1 rows (no WMMA in this kernel)
    float4 row[12];
    if (n < N) {
        float* rowf = (float*)row;
#pragma unroll
        for (int b = 0; b < 16; ++b) {
            const long long o = ((long long)b * N + n) * 3;
            rowf[b * 3 + 0] = g[o + 0] - p[o + 0];
            rowf[b * 3 + 1] = g[o + 1] - p[o + 1];
            rowf[b * 3 + 2] = g[o + 2] - p[o + 2];
        }
    } else {
#pragma unroll
        for (int k = 0; k < 12; ++k) row[k] = make_float4(0.f, 0.f, 0.f, 0.f);
    }
    float4* dst = (float4*)(dT + (long long)n * RB);
#pragma unroll
    for (int k = 0; k < 12; ++k) dst[k] = row[k];
}

// ---------------------------------------------------------------------------
// Pass 1 (specialized B=16, M=9): one thread per point. Branchless gathers:
// invalid indices are redirected to the zero pad row, so all 8 neighbor rows
// (192B contiguous each) are fetched unconditionally and overlap in flight.
// ---------------------------------------------------------------------------
__global__ void mcl_gather_kernel(const float* __restrict__ dT,
                                  const int*   __restrict__ nb,
                                  const float* __restrict__ nn,
                                  float* __restrict__ partials, int N) {
    const int n = blockIdx.x * blockDim.x + threadIdx.x;
    float s = 0.0f;
    if (n < N) {
        const int*  row = nb + (long long)n * 9;
        const float nnv = nn[n];
        int idx[8];
#pragma unroll
        for (int j = 0; j < 8; ++j) {
            const unsigned ii = (unsigned)row[j + 1];
            idx[j] = (ii < (unsigned)N) ? (int)ii : N;   // pad row -> zeros
        }
        float4 acc[12];
        const float4* cp = (const float4*)(dT + (long long)n * RB);
#pragma unroll
        for (int k = 0; k < 12; ++k) {
            const float4 c = cp[k];
            acc[k] = make_float4(c.x * nnv, c.y * nnv, c.z * nnv, c.w * nnv);
        }
#pragma unroll 2
        for (int j = 0; j < 8; ++j) {
            const float4* qp = (const float4*)(dT + (long long)idx[j] * RB);
#pragma unroll
            for (int k = 0; k < 12; ++k) {
                const float4 q = qp[k];
                acc[k].x -= q.x; acc[k].y -= q.y;
                acc[k].z -= q.z; acc[k].w -= q.w;
            }
        }
#pragma unroll
        for (int k = 0; k < 12; ++k)
            s += (fabsf(acc[k].x) + fabsf(acc[k].y)) +
                 (fabsf(acc[k].z) + fabsf(acc[k].w));
    }
    // EXEC all-ones here (no early returns above).
    block_store_partial(s, &partials[blockIdx.x]);
}

// ---------------------------------------------------------------------------
// Generic fallback (any B/M, no workspace table): direct two-tensor gather.
// ---------------------------------------------------------------------------
__global__ void mcl_lap_direct_kernel(const float* __restrict__ gt,
                                      const float* __restrict__ pr,
                                      const int*   __restrict__ nb,
                                      const float* __restrict__ nn,
                                      float* __restrict__ partials,
                                      int N, int M, int B) {
    const int n = blockIdx.x * blockDim.x + threadIdx.x;
    float s = 0.0f;
    if (n < N) {
        const int   m1  = M - 1;
        const int*  row = nb + (long long)n * M;
        const float nnv = nn[n];
        for (int b = 0; b < B; ++b) {
            const long long boff = (long long)b * N * 3;
            const float* g0 = gt + boff + (long long)n * 3;
            const float* p0 = pr + boff + (long long)n * 3;
            float a0 = (g0[0] - p0[0]) * nnv;
            float a1 = (g0[1] - p0[1]) * nnv;
            float a2 = (g0[2] - p0[2]) * nnv;
            for (int j = 1; j <= m1; ++j) {
                const int ii = row[j];
                if ((unsigned)ii < (unsigned)N) {
                    const float* gq = gt + boff + (long long)ii * 3;
                    const float* pq = pr + boff + (long long)ii * 3;
                    a0 -= gq[0] - pq[0];
                    a1 -= gq[1] - pq[1];
                    a2 -= gq[2] - pq[2];
                }
            }
            s += fabsf(a0) + fabsf(a1) + fabsf(a2);
        }
    }
    block_store_partial(s, &partials[blockIdx.x]);
}

// ---------------------------------------------------------------------------
// Pass 2: deterministic single-block reduction of per-block partials.
// ---------------------------------------------------------------------------
__global__ void mcl_finalize_kernel(const float* __restrict__ partials,
                                    int n, float scale,
                                    float* __restrict__ out) {
    float s = 0.0f;
    for (int i = threadIdx.x; i < n; i += blockDim.x) s += partials[i];
    const float w = wave_sum_wmma(s);
    __shared__ float wsum[8];
    const int lane = threadIdx.x & 31;
    const int wid  = threadIdx.x >> 5;
    if (lane == 0) wsum[wid] = w;
    __syncthreads();
    if (threadIdx.x == 0) {
        const int nw = (int)(blockDim.x >> 5);
        float t = 0.0f;
        for (int i = 0; i < nw; ++i) t += wsum[i];
        out[0] = t * scale;
    }
}

// ---------------------------------------------------------------------------
extern "C" void kernel_launch(void* const* d_in, const int* in_sizes, int n_in,
                              void* d_out, int out_size, void* d_ws, size_t ws_size,
                              hipStream_t stream) {
    const float* gt = (const float*)d_in[0];
    const float* pr = (const float*)d_in[1];
    const int*   nb = (const int*)  d_in[2];
    const float* nn = (const float*)d_in[3];
    float*       out = (float*)d_out;

    const int N = in_sizes[3];                 // 100000
    const int M = in_sizes[2] / N;             // 9
    const int B = in_sizes[0] / (3 * N);       // 16
    const long long E = (long long)B * N * 3;  // 4.8M
    const float scale = 1.0f / (float)E;

    const int block = 256;
    const int grid1 = (N + block - 1) / block;

    const size_t tblBytes = (((size_t)(N + 1) * RB * sizeof(float)) + 255) &
                            ~(size_t)255;
    const size_t need = tblBytes + (size_t)grid1 * sizeof(float);

    if (B == 16 && M == 9 && ws_size >= need) {
        float* dT       = (float*)d_ws;
        float* partials = (float*)((char*)d_ws + tblBytes);
        const int gridT = (N + 1 + block - 1) / block;
        mcl_difft_kernel<<<gridT, block, 0, stream>>>(gt, pr, dT, N);
        mcl_gather_kernel<<<grid1, block, 0, stream>>>(dT, nb, nn, partials, N);
        mcl_finalize_kernel<<<1, 256, 0, stream>>>(partials, grid1, scale, out);
    } else {
        float* partials = (float*)d_ws;
        mcl_lap_direct_kernel<<<grid1, block, 0, stream>>>(gt, pr, nb, nn,
                                                           partials, N, M, B);
        mcl_finalize_kernel<<<1, 256, 0, stream>>>(partials, grid1, scale, out);
    }
}